// Encoder_91147795956509
// MI455X (gfx1250) — compile-verified
//
#include <hip/hip_runtime.h>
#include <hip/hip_bf16.h>
#include <stdint.h>

#define NB     16      // batch
#define NPOS   784     // positions (28*28)
#define NDIM   10000   // hypervector dimension
#define NLEV   256     // levels
#define BLK    256     // threads per block (8 waves of 32)

typedef _Float16 h4 __attribute__((ext_vector_type(4)));
typedef float v2f __attribute__((ext_vector_type(2)));
typedef float v8f __attribute__((ext_vector_type(8)));

#if __has_builtin(__builtin_amdgcn_wmma_f32_16x16x4_f32)
#define HAVE_WMMA_16X16X4 1
#else
#define HAVE_WMMA_16X16X4 0
#endif

// ---------------------------------------------------------------------------
// Kernel 1: quantize pixel intensities to level indices, stored transposed
// idxT[p][b] so the encode kernels read all 16 batch indices of a position
// as one 16-byte wave-broadcast load.
// ---------------------------------------------------------------------------
__global__ void hdc_quantize_kernel(const float* __restrict__ x,
                                    unsigned char* __restrict__ idxT) {
  int i = blockIdx.x * blockDim.x + threadIdx.x;
  if (i >= NB * NPOS) return;
  int b = i / NPOS;
  int p = i - b * NPOS;
  int q = (int)rintf(x[i] * 255.0f);   // round-half-even matches jnp.round
  q = q < 0 ? 0 : (q > 255 ? 255 : q);
  idxT[p * NB + b] = (unsigned char)q;
}

// ---------------------------------------------------------------------------
// Kernel 2: f32 -> f16 conversion (values are +-1, exact). Used for V and P.
// ---------------------------------------------------------------------------
__global__ void hdc_cvt_f16_kernel(const float* __restrict__ src,
                                   _Float16* __restrict__ dst, int n4) {
  int i = blockIdx.x * blockDim.x + threadIdx.x;
  if (i >= n4) return;
  float4 v = ((const float4*)src)[i];
  h4 h;
  h.x = (_Float16)v.x; h.y = (_Float16)v.y;
  h.z = (_Float16)v.z; h.w = (_Float16)v.w;
  ((h4*)dst)[i] = h;
}

// ===========================================================================
// FAST path: f16 gather-bind-bundle.
//   grid = (ceil(NDIM/128), 4).  LDS: V16 tile 64KB + reduction 8KB = 72KB
//   -> 4 blocks/WGP.  32 position-splits (4 blocks x 8 waves) of 25 positions.
//   Partial sums (exact integers in f16, |s|<=784<2048) -> f32 partial bufs.
// ===========================================================================
#define TILE_D  128
#define PCHUNK  ((NPOS + 31) / 32)   // 25
#define LDS_F16_BYTES (NLEV * TILE_D * 2 + NB * TILE_D * 4)  // 65536 + 8192

__global__ __launch_bounds__(BLK) void hdc_encode_f16_kernel(
    const _Float16* __restrict__ p16,       // (784, 10000) f16 +-1
    const _Float16* __restrict__ v16,       // (256, 10000) f16 +-1
    const unsigned char* __restrict__ idxT, // (784, 16) u8
    float* __restrict__ partial) {          // (4, 16, 10000) f32
  extern __shared__ char smem[];
  _Float16 (*Vt)[TILE_D] = (_Float16(*)[TILE_D])smem;               // [256][128] f16
  float (*red)[TILE_D] = (float(*)[TILE_D])(smem + NLEV * TILE_D * 2);

  const int tid = threadIdx.x;
  const int d0  = blockIdx.x * TILE_D;

  for (int i = tid; i < NB * TILE_D; i += BLK)
    (&red[0][0])[i] = 0.0f;

  // Async global->LDS copy of the 256 x 128 f16 value-table tile (64KB):
  // 4096 16-byte chunks, 16 per thread; no VGPR round trip, ASYNCcnt tracked.
  #pragma unroll 4
  for (int j = 0; j < (NLEV * TILE_D * 2 / 16) / BLK; ++j) {
    int c   = tid + BLK * j;
    int row = c >> 4;            // 16 chunks (128 f16) per row
    int col = (c & 15) * 8;
    int dd  = d0 + col;
    if (dd > NDIM - 8) dd = NDIM - 8;   // clamp keeps 16B alignment (10000%8==0)
    unsigned lds_off = (unsigned)(uintptr_t)&Vt[row][col];
    unsigned long long ga =
        (unsigned long long)(const void*)(v16 + (size_t)row * NDIM + dd);
    asm volatile("global_load_async_to_lds_b128 %0, %1, off"
                 :: "v"(lds_off), "v"(ga) : "memory");
  }
#if __has_builtin(__builtin_amdgcn_s_wait_asynccnt)
  __builtin_amdgcn_s_wait_asynccnt(0);
#else
  asm volatile("s_wait_asynccnt 0x0" ::: "memory");
#endif
  __syncthreads();

  const int lane_d = (tid & 31) * 4;   // 4 d per lane -> 128 d per wave
  const int wv     = tid >> 5;
  int dP = d0 + lane_d;
  if (dP > NDIM - 4) dP = NDIM - 4;    // 8B-aligned clamp; garbage lanes never stored

  h4 acc[NB];
  #pragma unroll
  for (int b = 0; b < NB; ++b) acc[b] = (h4)0;

  const uint4* __restrict__ idx4 = (const uint4*)idxT;
  const int si    = blockIdx.y * 8 + wv;           // split index 0..31
  const int p_beg = si * PCHUNK;
  const int p_end = (p_beg + PCHUNK < NPOS) ? p_beg + PCHUNK : NPOS;

  for (int p = p_beg; p < p_end; ++p) {
    uint4 iw = idx4[p];                                    // 16 level bytes
    const h4 pv = *(const h4*)(p16 + (size_t)p * NDIM + dP);
    if (p + 1 < p_end)
      __builtin_prefetch(p16 + (size_t)(p + 1) * NDIM + dP, 0, 1);
    const unsigned wbits[4] = {iw.x, iw.y, iw.z, iw.w};
    #pragma unroll
    for (int g = 0; g < 4; ++g) {
      #pragma unroll
      for (int k = 0; k < 4; ++k) {
        const int b = g * 4 + k;
        const unsigned row = (wbits[g] >> (8 * k)) & 0xffu;
        const h4 vv = *(const h4*)&Vt[row][lane_d];        // ds_load_b64, 1-pass
        acc[b] = acc[b] + vv * pv;                         // 2x v_pk_fma_f16
      }
    }
  }

  // Cross-wave reduction via native ds_add_f32 atomics
  #pragma unroll
  for (int b = 0; b < NB; ++b) {
    atomicAdd(&red[b][lane_d + 0], (float)acc[b].x);
    atomicAdd(&red[b][lane_d + 1], (float)acc[b].y);
    atomicAdd(&red[b][lane_d + 2], (float)acc[b].z);
    atomicAdd(&red[b][lane_d + 3], (float)acc[b].w);
  }
  __syncthreads();

  // Plain stores of this block's partial (no global atomics needed)
  const int dl = tid & (TILE_D - 1);
  const int bh = (tid >> 7) * (NB / 2);
  const int dd = d0 + dl;
  float* pslice = partial + (size_t)blockIdx.y * NB * NDIM;
  if (dd < NDIM) {
    #pragma unroll
    for (int k = 0; k < NB / 2; ++k)
      pslice[(size_t)(bh + k) * NDIM + dd] = red[bh + k][dl];
  }
}

// ---------------------------------------------------------------------------
// Epilogue: 4-way partial reduction + sign.
// The reduction s[i] = sum_{k<4} partial[k][i] IS a matrix op: D = ones(16x4)
// x B(4x16), exactly V_WMMA_F32_16X16X4_F32 (K=4 = #position-splits).
// B layout (4x16, rows striped across lanes per VGPR): lanes 0-15 supply
// K=0 (VGPR0) / K=1 (VGPR1), lanes 16-31 supply K=2 / K=3 -> each wave loads
// every input element exactly once; all D rows equal the column sums, lane
// n<16 reads D[0][n] from the first D VGPR and stores the sign.
// ---------------------------------------------------------------------------
__global__ __launch_bounds__(BLK) void hdc_sign_kernel(
    const float* __restrict__ partial, float* __restrict__ out) {
  const int NBD = NB * NDIM;
#if HAVE_WMMA_16X16X4
  const int lane = threadIdx.x & 31;
  const int wv   = threadIdx.x >> 5;
  const int tile = blockIdx.x * (BLK / 32) + wv;   // one 16-column tile per wave
  if (tile >= NBD / 16) return;                    // wave-uniform guard
  const int  n  = lane & 15;
  const bool hi = lane >= 16;
  const int  i  = tile * 16 + n;

  v2f a; a.x = 1.0f; a.y = 1.0f;                   // ones(16x4)
  v2f b;
  b.x = partial[(size_t)(hi ? 2 : 0) * NBD + i];   // B VGPR0: K=0 / K=2
  b.y = partial[(size_t)(hi ? 3 : 1) * NBD + i];   // B VGPR1: K=1 / K=3
  v8f c = {0.f, 0.f, 0.f, 0.f, 0.f, 0.f, 0.f, 0.f};
  v8f d = __builtin_amdgcn_wmma_f32_16x16x4_f32(
      /*neg_a=*/false, a, /*neg_b=*/false, b,
      /*c_mod=*/(short)0, c, /*reuse_a=*/false, /*reuse_b=*/false);
  if (!hi) out[i] = d[0] > 0.0f ? 1.0f : -1.0f;
#else
  int i = blockIdx.x * blockDim.x + threadIdx.x;
  if (i >= NBD) return;
  float s = partial[i] + partial[NBD + i] +
            partial[2 * NBD + i] + partial[3 * NBD + i];
  out[i] = s > 0.0f ? 1.0f : -1.0f;
#endif
}

// ===========================================================================
// FALLBACK path (small workspace): f32 single-pass kernel.
// ===========================================================================
#define LDS_F32_FLOATS (NLEV * TILE_D + NB * TILE_D)

__global__ __launch_bounds__(BLK) void hdc_encode_f32_kernel(
    const float* __restrict__ pos, const float* __restrict__ val,
    const unsigned char* __restrict__ idxT, float* __restrict__ out) {
  extern __shared__ float smemf[];
  float (*Vt)[TILE_D]  = (float(*)[TILE_D])smemf;
  float (*red)[TILE_D] = (float(*)[TILE_D])(smemf + NLEV * TILE_D);

  const int tid = threadIdx.x;
  const int d0  = blockIdx.x * TILE_D;

  for (int i = tid; i < NB * TILE_D; i += BLK)
    (&red[0][0])[i] = 0.0f;

  #pragma unroll 4
  for (int j = 0; j < (NLEV * TILE_D / 4) / BLK; ++j) {
    int c   = tid + BLK * j;
    int row = c >> 5;
    int col = (c & 31) * 4;
    int dd  = d0 + col;
    if (dd > NDIM - 4) dd = NDIM - 4;
    unsigned lds_off = (unsigned)(uintptr_t)&Vt[row][col];
    unsigned long long gaddr =
        (unsigned long long)(const void*)(val + (size_t)row * NDIM + dd);
    asm volatile("global_load_async_to_lds_b128 %0, %1, off"
                 :: "v"(lds_off), "v"(gaddr) : "memory");
  }
#if __has_builtin(__builtin_amdgcn_s_wait_asynccnt)
  __builtin_amdgcn_s_wait_asynccnt(0);
#else
  asm volatile("s_wait_asynccnt 0x0" ::: "memory");
#endif
  __syncthreads();

  const int lane_d = (tid & 31) * 4;
  const int wv     = tid >> 5;
  int dP = d0 + lane_d;
  if (dP > NDIM - 4) dP = NDIM - 4;

  float acc[NB][4];
  #pragma unroll
  for (int b = 0; b < NB; ++b)
    #pragma unroll
    for (int c = 0; c < 4; ++c) acc[b][c] = 0.0f;

  const uint4* __restrict__ idx4 = (const uint4*)idxT;
  const int p_beg = wv * (NPOS / 8);
  const int p_end = p_beg + (NPOS / 8);

  for (int p = p_beg; p < p_end; ++p) {
    uint4 iw = idx4[p];
    const float4 pv = *(const float4*)(pos + (size_t)p * NDIM + dP);
    if (p + 1 < p_end)
      __builtin_prefetch(pos + (size_t)(p + 1) * NDIM + dP, 0, 1);
    const unsigned wbits[4] = {iw.x, iw.y, iw.z, iw.w};
    #pragma unroll
    for (int g = 0; g < 4; ++g) {
      #pragma unroll
      for (int k = 0; k < 4; ++k) {
        const int b = g * 4 + k;
        const unsigned row = (wbits[g] >> (8 * k)) & 0xffu;
        const float4 vv = *(const float4*)&Vt[row][lane_d];
        acc[b][0] = fmaf(vv.x, pv.x, acc[b][0]);
        acc[b][1] = fmaf(vv.y, pv.y, acc[b][1]);
        acc[b][2] = fmaf(vv.z, pv.z, acc[b][2]);
        acc[b][3] = fmaf(vv.w, pv.w, acc[b][3]);
      }
    }
  }

  #pragma unroll
  for (int b = 0; b < NB; ++b) {
    atomicAdd(&red[b][lane_d + 0], acc[b][0]);
    atomicAdd(&red[b][lane_d + 1], acc[b][1]);
    atomicAdd(&red[b][lane_d + 2], acc[b][2]);
    atomicAdd(&red[b][lane_d + 3], acc[b][3]);
  }
  __syncthreads();

  const int dl = tid & (TILE_D - 1);
  const int bh = (tid >> 7) * (NB / 2);
  const int dd = d0 + dl;
  if (dd < NDIM) {
    #pragma unroll
    for (int k = 0; k < NB / 2; ++k) {
      float s = red[bh + k][dl];
      out[(size_t)(bh + k) * NDIM + dd] = s > 0.0f ? 1.0f : -1.0f;
    }
  }
}

// ---------------------------------------------------------------------------
extern "C" void kernel_launch(void* const* d_in, const int* in_sizes, int n_in,
                              void* d_out, int out_size, void* d_ws, size_t ws_size,
                              hipStream_t stream) {
  (void)in_sizes; (void)n_in; (void)out_size;
  const float* x   = (const float*)d_in[0];  // (16,28,28)
  const float* pos = (const float*)d_in[1];  // (784,10000)
  const float* val = (const float*)d_in[2];  // (256,10000)
  float* out = (float*)d_out;                // (16,10000)

  // Workspace layout
  const size_t IDX_OFF  = 0;                               // 12544 B
  const size_t V16_OFF  = 16384;
  const size_t V16_SZ   = (size_t)NLEV * NDIM * 2;         // 5,120,000 B
  const size_t P16_OFF  = V16_OFF + V16_SZ;                // 64B aligned
  const size_t P16_SZ   = (size_t)NPOS * NDIM * 2;         // 15,680,000 B
  const size_t PART_OFF = P16_OFF + P16_SZ;
  const size_t PART_SZ  = 4ull * NB * NDIM * 4;            // 2,560,000 B
  const size_t NEED     = PART_OFF + PART_SZ;              // ~22.3 MB

  unsigned char* idxT = (unsigned char*)d_ws + IDX_OFF;

  const int nq = NB * NPOS;
  hdc_quantize_kernel<<<(nq + BLK - 1) / BLK, BLK, 0, stream>>>(x, idxT);

  const int ndtiles = (NDIM + TILE_D - 1) / TILE_D;  // 79

  if (ws_size >= NEED) {
    _Float16* v16 = (_Float16*)((char*)d_ws + V16_OFF);
    _Float16* p16 = (_Float16*)((char*)d_ws + P16_OFF);
    float* part   = (float*)((char*)d_ws + PART_OFF);

    int nv4 = NLEV * NDIM / 4;
    hdc_cvt_f16_kernel<<<(nv4 + BLK - 1) / BLK, BLK, 0, stream>>>(val, v16, nv4);
    int np4 = NPOS * NDIM / 4;
    hdc_cvt_f16_kernel<<<(np4 + BLK - 1) / BLK, BLK, 0, stream>>>(pos, p16, np4);

    dim3 grid(ndtiles, 4);
    hdc_encode_f16_kernel<<<grid, BLK, LDS_F16_BYTES, stream>>>(p16, v16, idxT, part);

#if HAVE_WMMA_16X16X4
    const int ntiles16 = NB * NDIM / 16;                       // 10000
    const int nsb = (ntiles16 + (BLK / 32) - 1) / (BLK / 32);  // 1250
#else
    const int nsb = (NB * NDIM + BLK - 1) / BLK;
#endif
    hdc_sign_kernel<<<nsb, BLK, 0, stream>>>(part, out);
  } else {
    hdc_encode_f32_kernel<<<ndtiles, BLK, LDS_F32_FLOATS * sizeof(float), stream>>>(
        pos, val, idxT, out);
  }
}